// HCCF_6725918785969
// MI455X (gfx1250) — compile-verified
//
#include <hip/hip_runtime.h>
#include <math.h>

// ---------------------------------------------------------------------------
// HCCF forward on MI455X (gfx1250), fp32 end-to-end.
// Dense GEMMs use V_WMMA_F32_16X16X4_F32 with 16x128 register blocking per
// wave (8 f32 accumulators), v2f (b64) fragment loads, and LDS-staged
// transposed B operands (ds_load_b64, conflict-free stride 132).
// SpMM is atomic-scatter (bandwidth bound).
// ---------------------------------------------------------------------------

#define N_USERS 100000
#define N_ITEMS 50000
#define DIM     128
#define NEDGE   1000000
#define BATCH   4096
#define SELSZ   (BATCH * DIM)
#define LDS_S   132  // padded LDS row stride (dwords): 4L+{0..3} banks, no conflicts

typedef float v2f __attribute__((ext_vector_type(2)));
typedef float v8f __attribute__((ext_vector_type(8)));

__device__ __forceinline__ float leakyf(float x) { return fmaxf(0.5f * x, x); }

__device__ __forceinline__ v8f wmma4(v2f a, v2f b, v8f c) {
  // D = A(16x4) * B(4x16) + C(16x16), fp32
  return __builtin_amdgcn_wmma_f32_16x16x4_f32(false, a, false, b, (short)0, c,
                                               false, false);
}

// Stage a 128x128 row-major matrix into LDS transposed: sB[n*LDS_S + k] = B[k*128 + n]
__device__ __forceinline__ void stage_bt(float* sB, const float* __restrict__ B) {
  for (int idx = threadIdx.x; idx < 128 * 128; idx += 256) {
    int k = idx >> 7, n = idx & 127;
    sB[n * LDS_S + k] = B[idx];
  }
  __syncthreads();
}

// ------------------------------- utility kernels ---------------------------

__global__ void zero_kernel(float* p, size_t n) {
  size_t i  = (size_t)blockIdx.x * blockDim.x + threadIdx.x;
  size_t st = (size_t)gridDim.x * blockDim.x;
  for (; i < n; i += st) p[i] = 0.0f;
}

__global__ void leaky_ip_kernel(float* p, int n) {
  int i = blockIdx.x * blockDim.x + threadIdx.x;
  if (i < n) p[i] = leakyf(p[i]);
}

__global__ void sumsq_kernel(const float* __restrict__ x, size_t n, float* out) {
  __shared__ float sm[256];
  size_t i  = (size_t)blockIdx.x * blockDim.x + threadIdx.x;
  size_t st = (size_t)gridDim.x * blockDim.x;
  float s = 0.0f;
  for (; i < n; i += st) { float v = x[i]; s += v * v; }
  sm[threadIdx.x] = s;
  __syncthreads();
  for (int o = 128; o > 0; o >>= 1) {
    if ((int)threadIdx.x < o) sm[threadIdx.x] += sm[threadIdx.x + o];
    __syncthreads();
  }
  if (threadIdx.x == 0) atomicAdd(out, sm[0]);
}

// ------------------------------- SpMM (COO scatter) ------------------------
// wave = one edge, lane = one float4 chunk of the 128-dim row. Does both
// tmpU[row] += v*icur[col]  and  tmpI[col] += v*ucur[row] in one pass.

__global__ void spmm_kernel(const int* __restrict__ rows,
                            const int* __restrict__ cols,
                            const float* __restrict__ vals,
                            const float* __restrict__ ucur,
                            const float* __restrict__ icur,
                            float* tmpU, float* tmpI) {
  long tid = (long)blockIdx.x * blockDim.x + threadIdx.x;
  if (tid >= (long)NEDGE * 32) return;
  int e = (int)(tid >> 5);
  int g = (int)(tid & 31);
  int r = rows[e], c = cols[e];
  float v = vals[e];
  float4 iv = ((const float4*)icur)[(size_t)c * 32 + g];
  float4 uv = ((const float4*)ucur)[(size_t)r * 32 + g];
  float* pu = tmpU + (size_t)r * DIM + g * 4;
  atomicAdd(pu + 0, v * iv.x);
  atomicAdd(pu + 1, v * iv.y);
  atomicAdd(pu + 2, v * iv.z);
  atomicAdd(pu + 3, v * iv.w);
  float* pi = tmpI + (size_t)c * DIM + g * 4;
  atomicAdd(pi + 0, v * uv.x);
  atomicAdd(pi + 1, v * uv.y);
  atomicAdd(pi + 2, v * uv.z);
  atomicAdd(pi + 3, v * uv.w);
}

// ------------------------------- WMMA GEMMs --------------------------------
// Fragment layout per CDNA5 ISA 7.12.2:
//  A 16x4 : lanes 0-15 -> M=lane, K={k0,k0+1}; lanes 16-31 -> M=lane-16, K={k0+2,k0+3}
//  C 16x16: vgpr r, lanes 0-15 -> (M=r, N=lane); lanes 16-31 -> (M=r+8, N=lane-16)

// C[M,128] = A[M,128] @ B[128,128] (optional leaky). B staged transposed in
// LDS; one wave computes a 16x128 strip (8 accumulators, A frag reused 8x).
__global__ void gemm_nn128_kernel(const float* __restrict__ A,
                                  const float* __restrict__ B, float* C, int M,
                                  int doLeaky) {
  __shared__ float sB[128 * LDS_S];
  stage_bt(sB, B);
  int wid  = (blockIdx.x * blockDim.x + threadIdx.x) >> 5;
  int lane = threadIdx.x & 31;
  if (wid >= (M >> 4)) return;  // wave-uniform (after barrier)
  int m0 = wid << 4;
  int half = lane >> 4, lq = lane & 15;
  v8f zero8 = {0.f, 0.f, 0.f, 0.f, 0.f, 0.f, 0.f, 0.f};
  v8f acc[8];
#pragma unroll
  for (int j = 0; j < 8; j++) acc[j] = zero8;
  for (int k0 = 0; k0 < DIM; k0 += 4) {
    v2f a = *(const v2f*)(A + (size_t)(m0 + lq) * DIM + k0 + 2 * half);
#pragma unroll
    for (int j = 0; j < 8; j++) {
      v2f b = *(const v2f*)(&sB[(j * 16 + lq) * LDS_S + k0 + 2 * half]);
      acc[j] = wmma4(a, b, acc[j]);
    }
  }
#pragma unroll
  for (int rr = 0; rr < 8; rr++) {
    int row = m0 + rr + 8 * half;
#pragma unroll
    for (int j = 0; j < 8; j++) {
      float v = acc[j][rr];
      if (doLeaky) v = leakyf(v);
      C[(size_t)row * DIM + j * 16 + lq] = v;
    }
  }
}

// C[128,128] += A^T @ B with A,B = [Krows,128]; split-K, atomicAdd into C.
// One wave computes a 16x128 strip of C over one K chunk.
__global__ void gemm_atb_split_kernel(const float* __restrict__ A,
                                      const float* __restrict__ B, float* C,
                                      int Krows, int chunk) {
  int wid  = (blockIdx.x * blockDim.x + threadIdx.x) >> 5;
  int lane = threadIdx.x & 31;
  int nch = (Krows + chunk - 1) / chunk;
  if (wid >= 8 * nch) return;
  int cidx = wid >> 3, mt = wid & 7;
  int m0 = mt << 4;
  int half = lane >> 4, lq = lane & 15;
  int kb = cidx * chunk;
  int ke = min(kb + chunk, Krows);
  v8f zero8 = {0.f, 0.f, 0.f, 0.f, 0.f, 0.f, 0.f, 0.f};
  v8f acc[8];
#pragma unroll
  for (int j = 0; j < 8; j++) acc[j] = zero8;
  for (int k0 = kb; k0 < ke; k0 += 4) {
    v2f a;  // A^T fragment (column of A, stride DIM)
    a.x = A[(size_t)(k0 + 2 * half) * DIM + m0 + lq];
    a.y = A[(size_t)(k0 + 2 * half + 1) * DIM + m0 + lq];
#pragma unroll
    for (int j = 0; j < 8; j++) {
      v2f b;
      b.x = B[(size_t)(k0 + 2 * half) * DIM + j * 16 + lq];
      b.y = B[(size_t)(k0 + 2 * half + 1) * DIM + j * 16 + lq];
      acc[j] = wmma4(a, b, acc[j]);
    }
  }
#pragma unroll
  for (int rr = 0; rr < 8; rr++)
#pragma unroll
    for (int j = 0; j < 8; j++)
      atomicAdd(&C[(size_t)(m0 + rr + 8 * half) * DIM + j * 16 + lq],
                acc[j][rr]);
}

// Lout = leaky(W^T @ Lin) + Lin; all 128x128. Single block of 8 waves,
// Lin staged transposed in LDS (also serves the residual read).
__global__ void hyper_mlp_kernel(const float* __restrict__ W,
                                 const float* __restrict__ Lin, float* Lout) {
  __shared__ float sB[128 * LDS_S];
  stage_bt(sB, Lin);
  int wid  = threadIdx.x >> 5;  // 8 waves, one 16-row strip each
  int lane = threadIdx.x & 31;
  int m0 = wid << 4;
  int half = lane >> 4, lq = lane & 15;
  v8f zero8 = {0.f, 0.f, 0.f, 0.f, 0.f, 0.f, 0.f, 0.f};
  v8f acc[8];
#pragma unroll
  for (int j = 0; j < 8; j++) acc[j] = zero8;
  for (int k0 = 0; k0 < DIM; k0 += 4) {
    v2f a;  // W^T fragment
    a.x = W[(size_t)(k0 + 2 * half) * DIM + m0 + lq];
    a.y = W[(size_t)(k0 + 2 * half + 1) * DIM + m0 + lq];
#pragma unroll
    for (int j = 0; j < 8; j++) {
      v2f b = *(const v2f*)(&sB[(j * 16 + lq) * LDS_S + k0 + 2 * half]);
      acc[j] = wmma4(a, b, acc[j]);
    }
  }
#pragma unroll
  for (int rr = 0; rr < 8; rr++) {
    int row = m0 + rr + 8 * half;
#pragma unroll
    for (int j = 0; j < 8; j++) {
      int col = j * 16 + lq;
      // residual Lin[row][col] from the transposed LDS copy
      Lout[(size_t)row * DIM + col] =
          leakyf(acc[j][rr]) + sB[col * LDS_S + row];
    }
  }
}

// Fused layer combine: cur = leaky(tmp) + leaky(Hyp@latW) + cur; sum += cur.
// latW staged transposed in LDS; one wave per 16-row strip.
__global__ void combine_kernel(const float* __restrict__ Hyp,
                               const float* __restrict__ latW,
                               const float* __restrict__ tmp, float* cur,
                               float* sum, int M) {
  __shared__ float sB[128 * LDS_S];
  stage_bt(sB, latW);
  int wid  = (blockIdx.x * blockDim.x + threadIdx.x) >> 5;
  int lane = threadIdx.x & 31;
  if (wid >= (M >> 4)) return;
  int m0 = wid << 4;
  int half = lane >> 4, lq = lane & 15;
  v8f zero8 = {0.f, 0.f, 0.f, 0.f, 0.f, 0.f, 0.f, 0.f};
  v8f acc[8];
#pragma unroll
  for (int j = 0; j < 8; j++) acc[j] = zero8;
  for (int k0 = 0; k0 < DIM; k0 += 4) {
    v2f a = *(const v2f*)(Hyp + (size_t)(m0 + lq) * DIM + k0 + 2 * half);
#pragma unroll
    for (int j = 0; j < 8; j++) {
      v2f b = *(const v2f*)(&sB[(j * 16 + lq) * LDS_S + k0 + 2 * half]);
      acc[j] = wmma4(a, b, acc[j]);
    }
  }
#pragma unroll
  for (int rr = 0; rr < 8; rr++) {
    int row = m0 + rr + 8 * half;
#pragma unroll
    for (int j = 0; j < 8; j++) {
      size_t idx = (size_t)row * DIM + j * 16 + lq;
      float nc = leakyf(acc[j][rr]) + leakyf(tmp[idx]) + cur[idx];
      cur[idx] = nc;
      sum[idx] += nc;
    }
  }
}

// Gather SSL slices at batch ids: gnnSel = leaky(tmp[id]); hypSel = leaky(Hyp[id]@latW)
__global__ void gather_sel_kernel(const int* __restrict__ ids,
                                  const float* __restrict__ Hyp,
                                  const float* __restrict__ latW,
                                  const float* __restrict__ tmp, float* gnnSel,
                                  float* hypSel) {
  __shared__ float sB[128 * LDS_S];
  stage_bt(sB, latW);
  int wid  = (blockIdx.x * blockDim.x + threadIdx.x) >> 5;
  int lane = threadIdx.x & 31;
  if (wid >= (BATCH >> 4)) return;
  int t0 = wid << 4;
  int half = lane >> 4, lq = lane & 15;
  int arow = ids[t0 + lq];  // row this lane supplies to the A fragment
  v8f zero8 = {0.f, 0.f, 0.f, 0.f, 0.f, 0.f, 0.f, 0.f};
  v8f acc[8];
#pragma unroll
  for (int j = 0; j < 8; j++) acc[j] = zero8;
  for (int k0 = 0; k0 < DIM; k0 += 4) {
    v2f a = *(const v2f*)(Hyp + (size_t)arow * DIM + k0 + 2 * half);
#pragma unroll
    for (int j = 0; j < 8; j++) {
      v2f b = *(const v2f*)(&sB[(j * 16 + lq) * LDS_S + k0 + 2 * half]);
      acc[j] = wmma4(a, b, acc[j]);
    }
  }
#pragma unroll
  for (int rr = 0; rr < 8; rr++) {
    int tr = t0 + rr + 8 * half;
    int srow = ids[tr];
#pragma unroll
    for (int j = 0; j < 8; j++) {
      int col = j * 16 + lq;
      hypSel[(size_t)tr * DIM + col] = leakyf(acc[j][rr]);
      gnnSel[(size_t)tr * DIM + col] = leakyf(tmp[(size_t)srow * DIM + col]);
    }
  }
}

// ------------------------------- SSL helpers -------------------------------

__global__ void l2norm_kernel(float* X, int rows) {
  int wid  = (blockIdx.x * blockDim.x + threadIdx.x) >> 5;
  int lane = threadIdx.x & 31;
  if (wid >= rows) return;
  float4* p = (float4*)(X + (size_t)wid * DIM) + lane;
  float4 v = *p;
  float s = v.x * v.x + v.y * v.y + v.z * v.z + v.w * v.w;
  for (int o = 16; o > 0; o >>= 1) s += __shfl_xor(s, o, 32);
  float inv = 1.0f / fmaxf(sqrtf(s), 1e-12f);
  v.x *= inv; v.y *= inv; v.z *= inv; v.w *= inv;
  *p = v;
}

__global__ void rowdot_kernel(const float* __restrict__ A,
                              const float* __restrict__ G, float* out,
                              int rows) {
  int wid  = (blockIdx.x * blockDim.x + threadIdx.x) >> 5;
  int lane = threadIdx.x & 31;
  if (wid >= rows) return;
  float4 a = ((const float4*)(A + (size_t)wid * DIM))[lane];
  float4 g = ((const float4*)(G + (size_t)wid * DIM))[lane];
  float s = a.x * g.x + a.y * g.y + a.z * g.z + a.w * g.w;
  for (int o = 16; o > 0; o >>= 1) s += __shfl_down(s, o, 32);
  if (lane == 0) out[wid] = s;
}

// neg[i] = sum_j exp(dot(G_i, A_j)): WMMA tiles of G @ A^T, one wave does a
// 16x128 strip (8 tiles) and pre-reduces exp() across tiles before atomics.
__global__ void neg_exp_kernel(const float* __restrict__ G,
                               const float* __restrict__ A, float* neg) {
  int wid  = (blockIdx.x * blockDim.x + threadIdx.x) >> 5;
  int lane = threadIdx.x & 31;
  const int NGRP = BATCH / 128;  // 32 column groups of 128
  if (wid >= (BATCH >> 4) * NGRP) return;
  int mt = wid / NGRP, ng = wid % NGRP;
  int m0 = mt << 4, n0 = ng << 7;
  int half = lane >> 4, lq = lane & 15;
  v8f zero8 = {0.f, 0.f, 0.f, 0.f, 0.f, 0.f, 0.f, 0.f};
  v8f acc[8];
#pragma unroll
  for (int j = 0; j < 8; j++) acc[j] = zero8;
  for (int k0 = 0; k0 < DIM; k0 += 4) {
    v2f a = *(const v2f*)(G + (size_t)(m0 + lq) * DIM + k0 + 2 * half);
#pragma unroll
    for (int j = 0; j < 8; j++) {
      // B = A^T fragment: contiguous along k in row-major A
      v2f b = *(const v2f*)(A + (size_t)(n0 + j * 16 + lq) * DIM + k0 + 2 * half);
      acc[j] = wmma4(a, b, acc[j]);
    }
  }
#pragma unroll
  for (int rr = 0; rr < 8; rr++) {
    float s = 0.0f;
#pragma unroll
    for (int j = 0; j < 8; j++) s += expf(acc[j][rr]);
    atomicAdd(&neg[m0 + rr + 8 * half], s);
  }
}

__global__ void ssl_final_kernel(const float* __restrict__ pos,
                                 const float* __restrict__ neg, float* out) {
  int i = blockIdx.x * blockDim.x + threadIdx.x;
  if (i >= BATCH) return;
  float v = -logf(expf(pos[i]) / (neg[i] + 1e-8f) + 1e-8f);
  atomicAdd(out, v);
}

__global__ void preds_kernel(const int* __restrict__ uids,
                             const int* __restrict__ iids,
                             const float* __restrict__ usum,
                             const float* __restrict__ isum, float* out) {
  int wid  = (blockIdx.x * blockDim.x + threadIdx.x) >> 5;
  int lane = threadIdx.x & 31;
  if (wid >= BATCH) return;
  int u = uids[wid], it = iids[wid];
  float4 a = ((const float4*)(usum + (size_t)u * DIM))[lane];
  float4 b = ((const float4*)(isum + (size_t)it * DIM))[lane];
  float s = a.x * b.x + a.y * b.y + a.z * b.z + a.w * b.w;
  for (int o = 16; o > 0; o >>= 1) s += __shfl_down(s, o, 32);
  if (lane == 0) out[wid] = s;
}

// ------------------------------- host driver -------------------------------

static inline dim3 wblocks(long waves) {
  return dim3((unsigned)((waves * 32 + 255) / 256));
}

extern "C" void kernel_launch(void* const* d_in, const int* in_sizes, int n_in,
                              void* d_out, int out_size, void* d_ws,
                              size_t ws_size, hipStream_t stream) {
  (void)in_sizes; (void)n_in; (void)out_size; (void)ws_size;
  const int*   uids   = (const int*)d_in[0];
  const int*   iids   = (const int*)d_in[1];
  const int*   arows  = (const int*)d_in[2];
  const int*   acols  = (const int*)d_in[3];
  const float* avals  = (const float*)d_in[4];
  const float* uE     = (const float*)d_in[5];
  const float* iE     = (const float*)d_in[6];
  const float* uhyper = (const float*)d_in[7];
  const float* ihyper = (const float*)d_in[8];
  const float* Wu     = (const float*)d_in[9];
  const float* Wi     = (const float*)d_in[10];
  const float* Wt     = (const float*)d_in[11];
  float* outp = (float*)d_out;

  float* w = (float*)d_ws;
  size_t o = 0;
  float* uuH   = w + o; o += (size_t)N_USERS * DIM;
  float* iiH   = w + o; o += (size_t)N_ITEMS * DIM;
  float* ucur  = w + o; o += (size_t)N_USERS * DIM;
  float* usum  = w + o; o += (size_t)N_USERS * DIM;
  float* tmpU  = w + o; o += (size_t)N_USERS * DIM;
  float* icur  = w + o; o += (size_t)N_ITEMS * DIM;
  float* isum  = w + o; o += (size_t)N_ITEMS * DIM;
  float* tmpI  = w + o; o += (size_t)N_ITEMS * DIM;
  float* latUA = w + o; o += DIM * DIM;
  float* latUB = w + o; o += DIM * DIM;
  float* latIA = w + o; o += DIM * DIM;
  float* latIB = w + o; o += DIM * DIM;
  float* gnnUsel = w + o; o += 2 * (size_t)SELSZ;
  float* hypUsel = w + o; o += 2 * (size_t)SELSZ;
  float* gnnIsel = w + o; o += 2 * (size_t)SELSZ;
  float* hypIsel = w + o; o += 2 * (size_t)SELSZ;
  float* Asel = w + o; o += SELSZ;
  float* negb = w + o; o += BATCH;
  float* posb = w + o; o += BATCH;

  // d_out: [0..4095]=preds, [4096]=ssl (accum), [4097]=reg (accum)
  zero_kernel<<<1, 64, 0, stream>>>(outp + BATCH, 2);

  // ulats[0]=uEmbed0 (cur and running sum); same for items
  hipMemcpyAsync(ucur, uE, (size_t)N_USERS * DIM * 4, hipMemcpyDeviceToDevice, stream);
  hipMemcpyAsync(usum, uE, (size_t)N_USERS * DIM * 4, hipMemcpyDeviceToDevice, stream);
  hipMemcpyAsync(icur, iE, (size_t)N_ITEMS * DIM * 4, hipMemcpyDeviceToDevice, stream);
  hipMemcpyAsync(isum, iE, (size_t)N_ITEMS * DIM * 4, hipMemcpyDeviceToDevice, stream);

  // Hyper adjacency: uuH = uE @ uhyper, iiH = iE @ ihyper (WMMA, LDS-staged B)
  gemm_nn128_kernel<<<wblocks(N_USERS / 16), 256, 0, stream>>>(uE, uhyper, uuH,
                                                               N_USERS, 0);
  gemm_nn128_kernel<<<wblocks(N_ITEMS / 16), 256, 0, stream>>>(iE, ihyper, iiH,
                                                               N_ITEMS, 0);

  // Regularizer
  sumsq_kernel<<<1024, 256, 0, stream>>>(uE, (size_t)N_USERS * DIM, outp + BATCH + 1);
  sumsq_kernel<<<1024, 256, 0, stream>>>(iE, (size_t)N_ITEMS * DIM, outp + BATCH + 1);
  sumsq_kernel<<<64, 256, 0, stream>>>(uhyper, (size_t)DIM * DIM, outp + BATCH + 1);
  sumsq_kernel<<<64, 256, 0, stream>>>(ihyper, (size_t)DIM * DIM, outp + BATCH + 1);

  for (int l = 0; l < 2; l++) {
    // SpMM both directions (uses pre-update cur on both sides)
    zero_kernel<<<4096, 256, 0, stream>>>(tmpU, (size_t)N_USERS * DIM);
    zero_kernel<<<2048, 256, 0, stream>>>(tmpI, (size_t)N_ITEMS * DIM);
    spmm_kernel<<<125000, 256, 0, stream>>>(arows, acols, avals, ucur, icur,
                                            tmpU, tmpI);

    // User hyper head: lat = leaky(uuH^T @ ucur); 3x (leaky(W^T lat)+lat)
    zero_kernel<<<16, 256, 0, stream>>>(latUA, (size_t)DIM * DIM);
    gemm_atb_split_kernel<<<wblocks(8L * (N_USERS / 5000)), 256, 0, stream>>>(
        uuH, ucur, latUA, N_USERS, 5000);
    leaky_ip_kernel<<<64, 256, 0, stream>>>(latUA, DIM * DIM);
    hyper_mlp_kernel<<<1, 256, 0, stream>>>(Wu + (size_t)(l * 3 + 0) * DIM * DIM, latUA, latUB);
    hyper_mlp_kernel<<<1, 256, 0, stream>>>(Wu + (size_t)(l * 3 + 1) * DIM * DIM, latUB, latUA);
    hyper_mlp_kernel<<<1, 256, 0, stream>>>(Wu + (size_t)(l * 3 + 2) * DIM * DIM, latUA, latUB);

    // Item hyper head
    zero_kernel<<<16, 256, 0, stream>>>(latIA, (size_t)DIM * DIM);
    gemm_atb_split_kernel<<<wblocks(8L * (N_ITEMS / 5000)), 256, 0, stream>>>(
        iiH, icur, latIA, N_ITEMS, 5000);
    leaky_ip_kernel<<<64, 256, 0, stream>>>(latIA, DIM * DIM);
    hyper_mlp_kernel<<<1, 256, 0, stream>>>(Wi + (size_t)(l * 3 + 0) * DIM * DIM, latIA, latIB);
    hyper_mlp_kernel<<<1, 256, 0, stream>>>(Wi + (size_t)(l * 3 + 1) * DIM * DIM, latIB, latIA);
    hyper_mlp_kernel<<<1, 256, 0, stream>>>(Wi + (size_t)(l * 3 + 2) * DIM * DIM, latIA, latIB);

    // SSL slice gathers (read tmp/Hyp/latW only; independent of cur update)
    gather_sel_kernel<<<wblocks(BATCH / 16), 256, 0, stream>>>(
        uids, uuH, latUB, tmpU, gnnUsel + (size_t)l * SELSZ, hypUsel + (size_t)l * SELSZ);
    gather_sel_kernel<<<wblocks(BATCH / 16), 256, 0, stream>>>(
        iids, iiH, latIB, tmpI, gnnIsel + (size_t)l * SELSZ, hypIsel + (size_t)l * SELSZ);

    // cur = leaky(tmp) + leaky(Hyp@latW) + cur ; sum += cur (fused WMMA)
    combine_kernel<<<wblocks(N_USERS / 16), 256, 0, stream>>>(uuH, latUB, tmpU,
                                                              ucur, usum, N_USERS);
    combine_kernel<<<wblocks(N_ITEMS / 16), 256, 0, stream>>>(iiH, latIB, tmpI,
                                                              icur, isum, N_ITEMS);
  }

  // preds = <usum[uid], isum[iid]>
  preds_kernel<<<wblocks(BATCH), 256, 0, stream>>>(uids, iids, usum, isum, outp);

  // SSL (permutation-invariant: sorted-unique == raw distinct ids)
  for (int l = 0; l < 2; l++) {
    for (int side = 0; side < 2; side++) {
      float* hyp = (side ? hypIsel : hypUsel) + (size_t)l * SELSZ;
      float* gnn = (side ? gnnIsel : gnnUsel) + (size_t)l * SELSZ;
      l2norm_kernel<<<wblocks(BATCH), 256, 0, stream>>>(hyp, BATCH);
      l2norm_kernel<<<wblocks(BATCH), 256, 0, stream>>>(gnn, BATCH);
      gemm_nn128_kernel<<<wblocks(BATCH / 16), 256, 0, stream>>>(
          hyp, Wt + (size_t)l * DIM * DIM, Asel, BATCH, 0);
      rowdot_kernel<<<wblocks(BATCH), 256, 0, stream>>>(Asel, gnn, posb, BATCH);
      zero_kernel<<<16, 256, 0, stream>>>(negb, BATCH);
      neg_exp_kernel<<<wblocks((long)(BATCH / 16) * (BATCH / 128)), 256, 0,
                       stream>>>(gnn, Asel, negb);
      ssl_final_kernel<<<16, 256, 0, stream>>>(posb, negb, outp + BATCH);
    }
  }
}